// GaInit_72009421685306
// MI455X (gfx1250) — compile-verified
//
#include <hip/hip_runtime.h>
#include <hip/hip_bf16.h>

// ---------------------------------------------------------------------------
// Gaussian OT map kernel for MI455X (gfx1250), wave32.
//
// Memory-bound problem: dominant traffic = read a,b (64MB) + write f (32MB)
// -> ~4.1us at 23.3 TB/s. Both streaming passes are expressed as skinny f32
// GEMMs on V_WMMA_F32_16X16X4_F32 (f32 required: weights ~6e-5).
//
//   1) feat_pair_kernel  : pair-interleaved feature table
//                          gt[j][c][p] = feat_c(point 2j+p), c in 0..7
//                          (feats: x0, x1, x0^2, x0*x1, x1^2, 0,0,0)
//   2) reduce_kernel     : S[b][c] = sum_n w[b,n]*feat_c(n) via WMMA,
//                          + sum w^2 in VALU; deterministic slab partials.
//   3) geo_mean_kernel   : unweighted feature means of geo_x.
//   4) solve_kernel      : closed-form 2x2 sqrtm/inv -> 6 poly coeffs/row,
//                          row-mean folded into constant term.
//   5) emit_kernel       : f[b,n] = C[b,:] . [1,x0,x1,x0^2,x0x1,x1^2] via
//                          2 WMMAs per 16x16 tile.
// ---------------------------------------------------------------------------

typedef __attribute__((ext_vector_type(2))) float v2f;
typedef __attribute__((ext_vector_type(8))) float v8f;

static constexpr int BB    = 512;        // batch rows
static constexpr int NN    = 16384;      // grid points
static constexpr int NSLAB = 32;         // K-split for pass-1 partials
static constexpr int SLAB  = NN / NSLAB; // 512
static constexpr int TPW   = 8;          // 16x16 tiles per wave in emit

// ---------------- 1) pair-interleaved feature table -------------------------
// gt[j*16 + c*2 + p] = feature c of point (2j+p).
__global__ __launch_bounds__(256) void feat_pair_kernel(
    const float* __restrict__ geo, float* __restrict__ gt) {
  int j = blockIdx.x * 256 + threadIdx.x;   // pair index
  if (j >= NN / 2) return;
  float x0e = geo[4 * j + 0], x1e = geo[4 * j + 1];
  float x0o = geo[4 * j + 2], x1o = geo[4 * j + 3];
  float* r = gt + (size_t)j * 16;
  r[0]  = x0e;        r[1]  = x0o;         // c0 = x0
  r[2]  = x1e;        r[3]  = x1o;         // c1 = x1
  r[4]  = x0e * x0e;  r[5]  = x0o * x0o;   // c2 = x0^2
  r[6]  = x0e * x1e;  r[7]  = x0o * x1o;   // c3 = x0*x1
  r[8]  = x1e * x1e;  r[9]  = x1o * x1o;   // c4 = x1^2
  r[10] = 0.f; r[11] = 0.f; r[12] = 0.f; r[13] = 0.f; r[14] = 0.f; r[15] = 0.f;
}

// ---------------- 2) weighted-moment reduction (WMMA f32 16x16x4) -----------
// grid = (BB/16, NSLAB), block = 32 (one wave).
// A 16x4 layout (ISA 7.12.2): lane l holds row l%16, K = 2*(l/16)+{0,1}.
// B 4x16 layout (assumed, mirrors C/D striping): VGPR v, lane l holds
//   B[2*(l/16)+v][l%16].
// Lanes 8..15 load duplicate feature columns (lrow&7): D columns 8..15 are
// garbage but finite and never stored — no masking/branching needed.
__global__ __launch_bounds__(32) void reduce_kernel(
    const float* __restrict__ w, const float* __restrict__ gt,
    float* __restrict__ partial, float* __restrict__ sqp) {
  const int lane = threadIdx.x;
  const int half = lane >> 4;
  const int lrow = lane & 15;
  const int b0   = blockIdx.x * 16;
  const int slab = blockIdx.y;
  const int k0   = slab * SLAB;

  const float* wp = w + (size_t)(b0 + lrow) * NN + (k0 + 2 * half);
  const float* gp = gt + (size_t)((k0 >> 1) + half) * 16 + (lane & 7) * 2;

  v8f acc0 = {0.f, 0.f, 0.f, 0.f, 0.f, 0.f, 0.f, 0.f};
  v8f acc1 = {0.f, 0.f, 0.f, 0.f, 0.f, 0.f, 0.f, 0.f};
  float sq0 = 0.f, sq1 = 0.f, sq2 = 0.f, sq3 = 0.f;

#pragma unroll 2
  for (int it = 0; it < SLAB / 8; ++it) {
    // chunk0: K = k..k+3 ; chunk1: K = k+4..k+7
    v2f aop0 = *(const v2f*)(wp);        // w[row][ka], w[row][ka+1]
    v2f aop1 = *(const v2f*)(wp + 4);
    v2f bop0 = *(const v2f*)(gp);        // {G[ka][c], G[ka+1][c]}
    v2f bop1 = *(const v2f*)(gp + 32);   // {G[ka+4][c], G[ka+5][c]}

    sq0 = fmaf(aop0.x, aop0.x, sq0);
    sq1 = fmaf(aop0.y, aop0.y, sq1);
    sq2 = fmaf(aop1.x, aop1.x, sq2);
    sq3 = fmaf(aop1.y, aop1.y, sq3);

    acc0 = __builtin_amdgcn_wmma_f32_16x16x4_f32(
        false, aop0, false, bop0, (short)0, acc0, false, false);
    acc1 = __builtin_amdgcn_wmma_f32_16x16x4_f32(
        false, aop1, false, bop1, (short)0, acc1, false, false);

    wp += 8;
    gp += 64;
  }

  v8f acc = acc0 + acc1;

  float sq = (sq0 + sq1) + (sq2 + sq3);
  sq += __shfl_xor(sq, 16, 32);   // combine the two K-halves (same row)

  // D layout: VGPR r, lanes 0-15 -> M=r; lanes 16-31 -> M=r+8; N = lrow.
  float* pbase = partial + ((size_t)slab * BB + b0) * 8;
  if (lrow < 8) {
#pragma unroll
    for (int r = 0; r < 8; ++r)
      pbase[(size_t)(r + 8 * half) * 8 + lrow] = acc[r];
  }
  if (lane < 16) sqp[(size_t)slab * BB + b0 + lane] = sq;
}

// ---------------- 3) unweighted geo_x feature means --------------------------
__global__ __launch_bounds__(256) void geo_mean_kernel(
    const float* __restrict__ geo, float* __restrict__ gbar) {
  __shared__ float sdat[5][256];
  float s[5] = {0.f, 0.f, 0.f, 0.f, 0.f};
  for (int i = threadIdx.x; i < NN; i += 256) {
    float x0 = geo[2 * i], x1 = geo[2 * i + 1];
    s[0] += x0; s[1] += x1; s[2] += x0 * x0; s[3] += x0 * x1; s[4] += x1 * x1;
  }
#pragma unroll
  for (int c = 0; c < 5; ++c) sdat[c][threadIdx.x] = s[c];
  __syncthreads();
  for (int stride = 128; stride > 0; stride >>= 1) {
    if (threadIdx.x < stride) {
#pragma unroll
      for (int c = 0; c < 5; ++c)
        sdat[c][threadIdx.x] += sdat[c][threadIdx.x + stride];
    }
    __syncthreads();
  }
  if (threadIdx.x == 0) {
#pragma unroll
    for (int c = 0; c < 5; ++c) gbar[c] = sdat[c][0] * (1.f / (float)NN);
  }
}

// ---------------- 4) per-batch closed-form 2x2 OT map -> poly coeffs --------
__global__ __launch_bounds__(256) void solve_kernel(
    const float* __restrict__ pA, const float* __restrict__ sqA,
    const float* __restrict__ pB, const float* __restrict__ sqB,
    const float* __restrict__ gbar, float* __restrict__ coeff) {
  int b = blockIdx.x * 256 + threadIdx.x;
  if (b >= BB) return;

  float SA[5] = {0, 0, 0, 0, 0}, SB[5] = {0, 0, 0, 0, 0};
  float qa = 0.f, qb = 0.f;
  for (int s = 0; s < NSLAB; ++s) {
    const float* pa = pA + ((size_t)s * BB + b) * 8;
    const float* pb = pB + ((size_t)s * BB + b) * 8;
#pragma unroll
    for (int c = 0; c < 5; ++c) { SA[c] += pa[c]; SB[c] += pb[c]; }
    qa += sqA[(size_t)s * BB + b];
    qb += sqB[(size_t)s * BB + b];
  }

  float ma0 = SA[0], ma1 = SA[1];
  float da = 1.f - qa;
  float ca00 = (SA[2] - ma0 * ma0) / da;
  float ca01 = (SA[3] - ma0 * ma1) / da;
  float ca11 = (SA[4] - ma1 * ma1) / da;

  float mb0 = SB[0], mb1 = SB[1];
  float db = 1.f - qb;
  float cb00 = (SB[2] - mb0 * mb0) / db;
  float cb01 = (SB[3] - mb0 * mb1) / db;
  float cb11 = (SB[4] - mb1 * mb1) / db;

  // sqrt0 = sqrtm(cov_a) = (M + s1*I)/t1 ; det(sqrt0) == s1
  float s1 = sqrtf(fmaxf(ca00 * ca11 - ca01 * ca01, 0.f));
  float t1 = sqrtf(fmaxf(ca00 + ca11 + 2.f * s1, 1e-30f));
  float r00 = (ca00 + s1) / t1, r01 = ca01 / t1, r11 = (ca11 + s1) / t1;
  float is = 1.f / fmaxf(s1, 1e-30f);
  float i00 = r11 * is, i01 = -r01 * is, i11 = r00 * is;   // sqrt0^{-1}

  // mid = sqrt0 * cov_b * sqrt0 (symmetric)
  float p00 = r00 * cb00 + r01 * cb01, p01 = r00 * cb01 + r01 * cb11;
  float p10 = r01 * cb00 + r11 * cb01, p11 = r01 * cb01 + r11 * cb11;
  float m00 = p00 * r00 + p01 * r01;
  float m01 = p00 * r01 + p01 * r11;
  float m11 = p10 * r01 + p11 * r11;

  // m = sqrtm(mid)
  float s2 = sqrtf(fmaxf(m00 * m11 - m01 * m01, 0.f));
  float t2 = sqrtf(fmaxf(m00 + m11 + 2.f * s2, 1e-30f));
  float q00 = (m00 + s2) / t2, q01 = m01 / t2, q11 = (m11 + s2) / t2;

  // A = sqrt0^{-1} * m * sqrt0^{-1} (symmetric)
  float u00 = i00 * q00 + i01 * q01, u01 = i00 * q01 + i01 * q11;
  float u10 = i01 * q00 + i11 * q01, u11 = i01 * q01 + i11 * q11;
  float A00 = u00 * i00 + u01 * i01;
  float A01 = u00 * i01 + u01 * i11;
  float A11 = u10 * i01 + u11 * i11;

  // f = |x|^2 - (x-ma)^T A (x-ma) - 2*mb.x   (polynomial in x0,x1)
  float c3 = 1.f - A00;                                   // x0^2
  float c5 = 1.f - A11;                                   // x1^2
  float c4 = -2.f * A01;                                  // x0*x1
  float c1 = 2.f * (A00 * ma0 + A01 * ma1) - 2.f * mb0;   // x0
  float c2 = 2.f * (A01 * ma0 + A11 * ma1) - 2.f * mb1;   // x1
  float c0 = -(A00 * ma0 * ma0 + 2.f * A01 * ma0 * ma1 + A11 * ma1 * ma1);

  // fold row-mean subtraction into the constant term
  float fmean = c0 + c1 * gbar[0] + c2 * gbar[1] + c3 * gbar[2] +
                c4 * gbar[3] + c5 * gbar[4];
  c0 -= fmean;

  float* cc = coeff + (size_t)b * 8;
  cc[0] = c0; cc[1] = c1; cc[2] = c2; cc[3] = c3;
  cc[4] = c4; cc[5] = c5; cc[6] = 0.f; cc[7] = 0.f;
}

// ---------------- 5) emit f = C x G^T via WMMA ------------------------------
__device__ __forceinline__ float featv(int k, float x0, float x1) {
  return (k == 0) ? 1.f
       : (k == 1) ? x0
       : (k == 2) ? x1
       : (k == 3) ? x0 * x0
       : (k == 4) ? x0 * x1
       : (k == 5) ? x1 * x1
                  : 0.f;
}

__global__ __launch_bounds__(256) void emit_kernel(
    const float* __restrict__ coeff, const float* __restrict__ geo,
    float* __restrict__ out) {
  const int lane = threadIdx.x & 31;
  const int wave = threadIdx.x >> 5;
  const int half = lane >> 4;
  const int lrow = lane & 15;
  const int b0 = blockIdx.x * 16;
  const int nbase = blockIdx.y * (8 * TPW * 16) + wave * (TPW * 16);

  // A operands: 16x4 coeff tiles (chunk0 = feats 0..3, chunk1 = feats 4..7)
  const float* crow = coeff + (size_t)(b0 + lrow) * 8;
  v2f a0 = *(const v2f*)(crow + 2 * half);
  v2f a1 = *(const v2f*)(crow + 4 + 2 * half);

  const int kb = 2 * half;
  const float* gpt = geo + 2 * (size_t)(nbase + lrow);
  float* obase = out + (size_t)(b0 + 8 * half) * NN + (nbase + lrow);

#pragma unroll
  for (int t = 0; t < TPW; ++t) {
    v2f g = *(const v2f*)gpt;            // (x0, x1) of this lane's point
    float x0 = g.x, x1 = g.y;
    v2f b0v, b1v;
    b0v.x = featv(kb,     x0, x1);  b0v.y = featv(kb + 1, x0, x1);
    b1v.x = featv(kb + 4, x0, x1);  b1v.y = featv(kb + 5, x0, x1);

    v8f acc = {0.f, 0.f, 0.f, 0.f, 0.f, 0.f, 0.f, 0.f};
    acc = __builtin_amdgcn_wmma_f32_16x16x4_f32(
        false, a0, false, b0v, (short)0, acc, false, false);
    acc = __builtin_amdgcn_wmma_f32_16x16x4_f32(
        false, a1, false, b1v, (short)0, acc, false, false);

    // D layout: VGPR r, lanes 0-15 -> row b0+r, lanes 16-31 -> row b0+r+8
#pragma unroll
    for (int r = 0; r < 8; ++r) obase[(size_t)r * NN] = acc[r];

    gpt += 32;     // next 16-point tile
    obase += 16;
  }
}

// ---------------------------------------------------------------------------
extern "C" void kernel_launch(void* const* d_in, const int* in_sizes, int n_in,
                              void* d_out, int out_size, void* d_ws,
                              size_t ws_size, hipStream_t stream) {
  const float* geo_x = (const float*)d_in[0];
  const float* geo_y = (const float*)d_in[1];
  const float* a     = (const float*)d_in[2];
  const float* b     = (const float*)d_in[3];
  float* out = (float*)d_out;

  // workspace layout (floats), ~2.25 MB total
  float* wsf   = (float*)d_ws;
  float* Gxt   = wsf;                            // NN*8  (pair-interleaved)
  float* Gyt   = Gxt + (size_t)NN * 8;           // NN*8
  float* pA    = Gyt + (size_t)NN * 8;           // NSLAB*BB*8
  float* pB    = pA + (size_t)NSLAB * BB * 8;    // NSLAB*BB*8
  float* sqA   = pB + (size_t)NSLAB * BB * 8;    // NSLAB*BB
  float* sqB   = sqA + (size_t)NSLAB * BB;       // NSLAB*BB
  float* gbar  = sqB + (size_t)NSLAB * BB;       // 8
  float* coeff = gbar + 8;                       // BB*8

  feat_pair_kernel<<<(NN / 2) / 256, 256, 0, stream>>>(geo_x, Gxt);
  feat_pair_kernel<<<(NN / 2) / 256, 256, 0, stream>>>(geo_y, Gyt);
  geo_mean_kernel<<<1, 256, 0, stream>>>(geo_x, gbar);

  reduce_kernel<<<dim3(BB / 16, NSLAB), 32, 0, stream>>>(a, Gxt, pA, sqA);
  reduce_kernel<<<dim3(BB / 16, NSLAB), 32, 0, stream>>>(b, Gyt, pB, sqB);

  solve_kernel<<<dim3(BB / 256), 256, 0, stream>>>(pA, sqA, pB, sqB, gbar,
                                                   coeff);

  emit_kernel<<<dim3(BB / 16, NN / (8 * TPW * 16)), 256, 0, stream>>>(
      coeff, geo_x, out);
}